// Rotation_20572893348086
// MI455X (gfx1250) — compile-verified
//
#include <hip/hip_runtime.h>

#define BN 65536
#define DN 512
#define RN 64
#define WPITCH 66           // sW pitch: even (8B-aligned v2f) + bank-spread
#define XPITCH 66           // TDM-padded x-chunk pitch: 64 dw row + 2 dw pad
#define CHUNK 64            // K columns staged per TDM transfer
#define NCHUNK (DN / CHUNK) // 8
#define XBUFB (128 * XPITCH * 4)  // bytes per x-chunk buffer (33792)

typedef __attribute__((ext_vector_type(2))) float v2f;
typedef __attribute__((ext_vector_type(8))) float v8f;
typedef __attribute__((ext_vector_type(4))) unsigned int v4u;
typedef __attribute__((ext_vector_type(8))) int v8i;
typedef __attribute__((ext_vector_type(4))) int v4i;

static __device__ __forceinline__ v8f wmma_f32(v2f a, v2f b, v8f c) {
  // V_WMMA_F32_16X16X4_F32 : D(16x16 f32) = A(16x4 f32) * B(4x16 f32) + C
  return __builtin_amdgcn_wmma_f32_16x16x4_f32(
      false, a, false, b, (short)0, c, false, false);
}

// TDM: stage a 128-row x 64-col f32 tile of x into LDS, padded +2 dwords per
// 64-dword row (pad_interval=64dw -> code 5, pad_amount=2dw -> code 1).
// D# per CDNA5 ISA ch.8: group0 = {flags, lds_addr, gaddr_lo, gaddr_hi|type2},
// group1 = dims/strides in data_size units. 2-D tensor -> groups 2/3 zero.
static __device__ __forceinline__ void tdm_load_chunk(const float* x,
                                                      int row_blk, int c,
                                                      unsigned lds_off) {
  unsigned long long ga =
      (unsigned long long)(uintptr_t)(x + (size_t)row_blk * DN + c * CHUNK);
  v4u g0;
  g0.x = 1u;                                          // count=1, user mode
  g0.y = lds_off;                                     // LDS byte address
  g0.z = (unsigned)(ga & 0xFFFFFFFFu);                // global_addr[31:0]
  g0.w = (unsigned)((ga >> 32) & 0x01FFFFFFu) | 0x80000000u; // [56:32]|type=2
  v8i g1;
  g1[0] = (2 << 16) | (1 << 20) | (5 << 22) | (1 << 25); // 4B elems, pad 64dw->+2dw
  g1[1] = (int)(512u << 16);                          // tensor_dim0 = 512
  g1[2] = 0;                                          // dim0 hi, dim1 lo16 (65536)
  g1[3] = (CHUNK << 16) | 1;                          // dim1 hi16=1, tile_dim0=64
  g1[4] = 128;                                        // tile_dim1 = 128 rows
  g1[5] = 512;                                        // tensor_dim0_stride = 512
  g1[6] = 0;
  g1[7] = 0;
  v4i z4 = {0, 0, 0, 0};
  v8i z8 = {0, 0, 0, 0, 0, 0, 0, 0};
  // 6-arg (clang-23 / therock) form: (g0, g1, g2, g3, g4, cpol)
  __builtin_amdgcn_tensor_load_to_lds(g0, g1, z4, z4, z8, 0);
}

// ---------------------------------------------------------------------------
// Prep: V = normalized columns, T = compact-WY factor (H1..HR = I - V T V^T),
// then pre-pack V and T into WMMA B-fragment order so the apply kernel's
// B operands are single coalesced b64 loads:
//   Vb [ks=0..127][nb=0..3][lane]  : B frag of V   (phase 1)
//   Tb [ks=0..15 ][nb=0..3][lane]  : B frag of T   (phase 2)
//   Vtb[nb=0..31 ][ks=0..15][lane] : B frag of V^T (phase 3)
// ---------------------------------------------------------------------------
__global__ __launch_bounds__(1024) void hh_prep(const float* __restrict__ P,
                                                float* __restrict__ V,
                                                float* __restrict__ T,
                                                float* __restrict__ G,
                                                float* __restrict__ Vb,
                                                float* __restrict__ Tb,
                                                float* __restrict__ Vtb) {
  __shared__ float sred[1024];
  __shared__ float snrm[RN];
  const int tid = threadIdx.x;
  const int col = tid >> 4;
  const int sub = tid & 15;

  float a = 0.f;
  for (int d = sub; d < DN; d += 16) {
    float v = P[d * RN + col];
    a += v * v;
  }
  sred[tid] = a;
  __syncthreads();
  if (sub == 0) {
    float s = 0.f;
    for (int j = 0; j < 16; ++j) s += sred[(col << 4) + j];
    snrm[col] = rsqrtf(s);
  }
  __syncthreads();

  for (int i = tid; i < DN * RN; i += 1024) V[i] = P[i] * snrm[i % RN];
  __threadfence();
  __syncthreads();

  for (int p = tid; p < RN * RN; p += 1024) {
    const int i = p / RN, k = p % RN;
    float s = 0.f;
    for (int d = 0; d < DN; ++d) s += V[d * RN + i] * V[d * RN + k];
    G[p] = s;
    T[p] = 0.f;
  }
  __threadfence();
  __syncthreads();
  if (tid == 0) T[0] = 2.f;
  __threadfence();
  __syncthreads();
  for (int k = 1; k < RN; ++k) {
    if (tid < k) {
      float s = 0.f;
      for (int j = tid; j < k; ++j) s += T[tid * RN + j] * G[j * RN + k];
      T[tid * RN + k] = -2.f * s;
    }
    if (tid == 0) T[k * RN + k] = 2.f;
    __threadfence();
    __syncthreads();
  }
  __threadfence();
  __syncthreads();

  // ---- pack B fragments (lane: h = lane>>4 selects K pair, m = lane&15 = N)
  for (int i = tid; i < 128 * 4 * 32; i += 1024) {   // Vb (phase 1)
    const int lane = i & 31, nb = (i >> 5) & 3, ks = i >> 7;
    const int h = lane >> 4, m = lane & 15;
    const int k = ks * 4 + 2 * h, n = nb * 16 + m;
    Vb[2 * i]     = V[k * RN + n];
    Vb[2 * i + 1] = V[(k + 1) * RN + n];
  }
  for (int i = tid; i < 16 * 4 * 32; i += 1024) {    // Tb (phase 2)
    const int lane = i & 31, nb = (i >> 5) & 3, ks = i >> 7;
    const int h = lane >> 4, m = lane & 15;
    const int k = ks * 4 + 2 * h, n = nb * 16 + m;
    Tb[2 * i]     = T[k * RN + n];
    Tb[2 * i + 1] = T[(k + 1) * RN + n];
  }
  for (int i = tid; i < 32 * 16 * 32; i += 1024) {   // Vtb (phase 3, V^T)
    const int lane = i & 31, ks = (i >> 5) & 15, nb = i >> 9;
    const int h = lane >> 4, m = lane & 15;
    const int k = ks * 4 + 2 * h, n = nb * 16 + m;
    Vtb[2 * i]     = V[n * RN + k];
    Vtb[2 * i + 1] = V[n * RN + k + 1];
  }
}

// ---------------------------------------------------------------------------
// Fused apply: out = x - ((x*V)*T)*V^T, one streaming pass.
// 256 thr = 8 waves; wave owns 16 rows; block = 128 rows; grid = 512 blocks.
// x staged into LDS by TDM (double-buffered, padded rows); all B operands are
// coalesced b64 loads of pre-packed fragments (L2-resident broadcasts).
// ---------------------------------------------------------------------------
__global__ __launch_bounds__(256) void hh_apply(const float* __restrict__ x,
                                                const float* __restrict__ Vb,
                                                const float* __restrict__ Tb,
                                                const float* __restrict__ Vtb,
                                                float* __restrict__ outx) {
  __shared__ float sX[2][128 * XPITCH];   // TDM destination, 2 x 33792 B
  __shared__ float sW[8][16 * WPITCH];    // per-wave 16x64 C<->A staging

  const int tid  = threadIdx.x;
  const int w    = tid >> 5;     // wave id (wave32)
  const int lane = tid & 31;
  const int half = lane >> 4;
  const int m    = lane & 15;

  const int row_blk = blockIdx.x * 128;
  const int row0 = row_blk + w * 16;
  const unsigned lds0 = (unsigned)(uintptr_t)(void*)&sX[0][0];

  const v2f* VbV  = (const v2f*)Vb;
  const v2f* TbV  = (const v2f*)Tb;
  const v2f* VtbV = (const v2f*)Vtb;
  float* Wm = &sW[w][0];

  // ---- Phase 1: W(16x64) = x_tile(16x512) * V(512x64), TDM-pipelined ------
  if (w == 0) {
    tdm_load_chunk(x, row_blk, 0, lds0);
    __builtin_amdgcn_s_wait_tensorcnt(0);
  }
  __syncthreads();

  v8f acc0 = {}, acc1 = {}, acc2 = {}, acc3 = {};
  for (int c = 0; c < NCHUNK; ++c) {
    if (w == 0 && c + 1 < NCHUNK)
      tdm_load_chunk(x, row_blk, c + 1, lds0 + (unsigned)(((c + 1) & 1) * XBUFB));

    const float* xc = &sX[c & 1][0];
#pragma unroll
    for (int kk = 0; kk < CHUNK; kk += 4) {
      const int ks = c * (CHUNK / 4) + (kk >> 2);
      v2f a = *(const v2f*)&xc[(w * 16 + m) * XPITCH + kk + 2 * half];
      const v2f* bb = VbV + (size_t)ks * 128 + lane;
      acc0 = wmma_f32(a, bb[0],  acc0);
      acc1 = wmma_f32(a, bb[32], acc1);
      acc2 = wmma_f32(a, bb[64], acc2);
      acc3 = wmma_f32(a, bb[96], acc3);
    }

    if (w == 0) __builtin_amdgcn_s_wait_tensorcnt(0);
    __syncthreads();
  }

#pragma unroll
  for (int r = 0; r < 8; ++r) {            // C layout -> row-major LDS
    const int row = r + 8 * half;
    Wm[row * WPITCH + m +  0] = acc0[r];
    Wm[row * WPITCH + m + 16] = acc1[r];
    Wm[row * WPITCH + m + 32] = acc2[r];
    Wm[row * WPITCH + m + 48] = acc3[r];
  }

  // ---- Phase 2: Z(16x64) = W * T ------------------------------------------
  v8f z0 = {}, z1 = {}, z2 = {}, z3 = {};
#pragma unroll
  for (int ks = 0; ks < 16; ++ks) {
    v2f a = *(const v2f*)&Wm[m * WPITCH + ks * 4 + 2 * half];
    const v2f* bb = TbV + (size_t)ks * 128 + lane;
    z0 = wmma_f32(a, bb[0],  z0);
    z1 = wmma_f32(a, bb[32], z1);
    z2 = wmma_f32(a, bb[64], z2);
    z3 = wmma_f32(a, bb[96], z3);
  }
#pragma unroll
  for (int r = 0; r < 8; ++r) {
    const int row = r + 8 * half;
    Wm[row * WPITCH + m +  0] = z0[r];
    Wm[row * WPITCH + m + 16] = z1[r];
    Wm[row * WPITCH + m + 32] = z2[r];
    Wm[row * WPITCH + m + 48] = z3[r];
  }

  // ---- Phase 3: out_tile = x_tile - Z(16x64) * V^T(64x512) ----------------
  for (int nb = 0; nb < DN / 16; ++nb) {
    const int n0 = nb * 16;
    v8f acc = {};
#pragma unroll
    for (int ks = 0; ks < 16; ++ks) {
      v2f a = *(const v2f*)&Wm[m * WPITCH + ks * 4 + 2 * half];
      v2f b = VtbV[((size_t)nb * 16 + ks) * 32 + lane];
      acc = wmma_f32(a, b, acc);
    }
#pragma unroll
    for (int r = 0; r < 8; ++r) {
      const size_t idx = (size_t)(row0 + r + 8 * half) * DN + n0 + m;
      outx[idx] = x[idx] - acc[r];   // re-read hits L2 (tile just DMA'd)
    }
  }
}

__global__ __launch_bounds__(256) void copy_sldj(const float* __restrict__ s,
                                                 float* __restrict__ o) {
  const int i = blockIdx.x * blockDim.x + threadIdx.x;
  o[i] = s[i];
}

extern "C" void kernel_launch(void* const* d_in, const int* in_sizes, int n_in,
                              void* d_out, int out_size, void* d_ws, size_t ws_size,
                              hipStream_t stream) {
  const float* x    = (const float*)d_in[0];
  const float* sldj = (const float*)d_in[1];
  const float* P    = (const float*)d_in[2];

  float* outx = (float*)d_out;
  float* outs = outx + (size_t)BN * DN;

  float* V   = (float*)d_ws;            // 32768
  float* T   = V   + DN * RN;           // 4096
  float* G   = T   + RN * RN;           // 4096
  float* Vb  = G   + RN * RN;           // 32768  (128*4*32 v2f)
  float* Tb  = Vb  + 128 * 4 * 32 * 2;  // 4096   (16*4*32 v2f)
  float* Vtb = Tb  + 16 * 4 * 32 * 2;   // 32768  (32*16*32 v2f)

  hh_prep<<<1, 1024, 0, stream>>>(P, V, T, G, Vb, Tb, Vtb);
  hh_apply<<<BN / 128, 256, 0, stream>>>(x, Vb, Tb, Vtb, outx);
  copy_sldj<<<BN / 256, 256, 0, stream>>>(sldj, outs);
}